// AttentionMechanism_63333587747046
// MI455X (gfx1250) — compile-verified
//
#include <hip/hip_runtime.h>

typedef __attribute__((ext_vector_type(2))) float v2f;
typedef __attribute__((ext_vector_type(8))) float v8f;

#define B_SZ 4096
#define DIN  400
#define KMIX 10
#define LSEQ 256
#define CVOC 80
#define NPAD 32   // W columns padded 30 -> 32 in LDS

// ---------------------------------------------------------------------------
// Kernel 1: raw = inputs @ W + b via V_WMMA_F32_16X16X4_F32, then activations.
//   alpha = exp(raw[:, 0:10])                        -> alpha_ws
//   beta  = exp(raw[:,10:20])                        -> beta_ws
//   kappa = prev_kappa + exp(raw[:,20:30] + kscale)  -> kappa_out (d_out)
// One block = 16 batch rows, 2 waves; wave w owns N-tile w (cols 16w..16w+15).
// W is staged once into LDS zero-padded to [400][32], so the 100-iteration
// WMMA loop is branch-free: global_load_b64 (A) + ds loads (B) + wmma.
// ---------------------------------------------------------------------------
__global__ __launch_bounds__(64) void proj_act_kernel(
    const float* __restrict__ inputs,        // [B, 400]
    const float* __restrict__ W,             // [400, 30]
    const float* __restrict__ bias,          // [30]
    const float* __restrict__ prev_kappa,    // [B, 10]
    const float* __restrict__ kappa_scale_p, // [1]
    float* __restrict__ alpha_ws,            // [B, 10]
    float* __restrict__ beta_ws,             // [B, 10]
    float* __restrict__ kappa_out)           // [B, 10]
{
    __shared__ float s_W[DIN * NPAD];        // 51,200 B, zero-padded weights

    // cooperative staging: [400][30] -> [400][32] with cols 30/31 = 0
    for (int i = threadIdx.x; i < DIN * NPAD; i += 64) {
        const int col = i & (NPAD - 1);
        const int row = i >> 5;
        s_W[i] = (col < 3 * KMIX) ? W[row * (3 * KMIX) + col] : 0.0f;
    }
    __syncthreads();

    const int lane = threadIdx.x & 31;
    const int wv   = threadIdx.x >> 5;   // N-tile: 0 or 1
    const int half = lane >> 4;          // lane group (K-offset / row+8 select)
    const int lm   = lane & 15;

    const int m0   = blockIdx.x * 16;
    const int colB = wv * 16 + lm;       // output column 0..31 (30/31 padded)

    // A-frag source: row (m0+lm), K elements {kb+2*half, kb+2*half+1}
    const float* arow = inputs + (size_t)(m0 + lm) * DIN + 2 * half;
    const float* brow = s_W + colB;

    v8f acc = {};
    for (int kb = 0; kb < DIN; kb += 4) {
        v2f a;
        a.x = arow[kb];
        a.y = arow[kb + 1];
        const int kr = kb + 2 * half;    // B rows {kr, kr+1}, column colB
        v2f bf;
        bf.x = brow[kr * NPAD];
        bf.y = brow[(kr + 1) * NPAD];
        acc = __builtin_amdgcn_wmma_f32_16x16x4_f32(
            /*neg_a=*/false, a, /*neg_b=*/false, bf,
            /*c_mod=*/(short)0, acc, /*reuse_a=*/false, /*reuse_b=*/false);
    }

    if (colB >= 3 * KMIX) return;        // divergence only after last WMMA
    const float ksc  = kappa_scale_p[0];
    const float bcol = bias[colB];
#pragma unroll
    for (int r = 0; r < 8; ++r) {
        const int row = m0 + r + 8 * half;   // C layout: M = r + 8*(lane>=16)
        const float val = acc[r] + bcol;
        if (colB < KMIX) {
            alpha_ws[row * KMIX + colB] = __expf(val);
        } else if (colB < 2 * KMIX) {
            beta_ws[row * KMIX + (colB - KMIX)] = __expf(val);
        } else {
            const int k = colB - 2 * KMIX;
            kappa_out[row * KMIX + k] =
                prev_kappa[row * KMIX + k] + __expf(val + ksc);
        }
    }
}

// ---------------------------------------------------------------------------
// Kernel 2: phi + masked one-hot einsum. One block per batch row.
//   phase 1: thread l computes phi[l] (10-term Gaussian mixture) -> LDS
//   phase 2: 240 threads = 12 l-phases x 20 float4 column groups stream the
//            [256,80] one-hot slab (coalesced b128 loads + global_prefetch),
//            accumulate phi-weighted columns, LDS reduce, float4 stores.
// This kernel carries the 335 MB / 23.3 TB/s ~ 14.4 us roofline.
// ---------------------------------------------------------------------------
__global__ __launch_bounds__(256) void phi_einsum_kernel(
    const float* __restrict__ alpha_ws,   // [B, 10]
    const float* __restrict__ beta_ws,    // [B, 10]
    const float* __restrict__ kappa,      // [B, 10] (from d_out)
    const float* __restrict__ onehot,     // [B, 256, 80]
    const int*   __restrict__ seqlen,     // [B]
    float* __restrict__ w_out)            // [B, 80]
{
    __shared__ float  s_alpha[KMIX];
    __shared__ float  s_beta[KMIX];
    __shared__ float  s_kappa[KMIX];
    __shared__ float  s_phi[LSEQ];
    __shared__ float4 s_part[12 * 20];

    const int b = blockIdx.x;
    const int t = threadIdx.x;

    if (t < KMIX) {
        s_alpha[t] = alpha_ws[b * KMIX + t];
        s_beta[t]  = beta_ws[b * KMIX + t];
        s_kappa[t] = kappa[b * KMIX + t];
    }
    __syncthreads();

    // phi[l], l = thread id
    {
        const int   l = t;
        const float u = (float)(l + 1);
        float s = 0.0f;
#pragma unroll
        for (int k = 0; k < KMIX; ++k) {
            const float d = s_kappa[k] - u;
            float e = -s_beta[k] * d * d;
            e = fminf(50.0f, fmaxf(-50.0f, e));
            s += s_alpha[k] * __expf(e);
        }
        if (l >= seqlen[b]) s = 0.0f;
        s_phi[l] = s;
    }
    __syncthreads();

    // streaming einsum over the one-hot slab
    if (t < 240) {
        const int c4   = t % 20;          // float4 column group: cols 4*c4..
        const int part = t / 20;          // l-phase 0..11
        const float4* oh =
            (const float4*)(onehot + (size_t)b * LSEQ * CVOC);
        float4 acc = make_float4(0.f, 0.f, 0.f, 0.f);
        for (int l = part; l < LSEQ; l += 12) {
            // speculative prefetch ~3 phases ahead (global_prefetch_b8)
            if (l + 36 < LSEQ)
                __builtin_prefetch(&oh[(l + 36) * (CVOC / 4) + c4], 0, 0);
            const float  p = s_phi[l];
            const float4 v = oh[l * (CVOC / 4) + c4];
            acc.x += p * v.x;
            acc.y += p * v.y;
            acc.z += p * v.z;
            acc.w += p * v.w;
        }
        s_part[part * 20 + c4] = acc;
    }
    __syncthreads();

    if (t < 20) {
        float4 sum = make_float4(0.f, 0.f, 0.f, 0.f);
#pragma unroll
        for (int p = 0; p < 12; ++p) {
            const float4 v = s_part[p * 20 + t];
            sum.x += v.x;
            sum.y += v.y;
            sum.z += v.z;
            sum.w += v.w;
        }
        ((float4*)(w_out + (size_t)b * CVOC))[t] = sum;
    }
}

// ---------------------------------------------------------------------------
extern "C" void kernel_launch(void* const* d_in, const int* in_sizes, int n_in,
                              void* d_out, int out_size, void* d_ws, size_t ws_size,
                              hipStream_t stream) {
    const float* inputs     = (const float*)d_in[0];  // [4096,400]
    const float* prev_kappa = (const float*)d_in[1];  // [4096,10]
    const float* onehot     = (const float*)d_in[2];  // [4096,256,80]
    const int*   seqlen     = (const int*)  d_in[3];  // [4096]
    const float* W          = (const float*)d_in[4];  // [400,30]
    const float* bias       = (const float*)d_in[5];  // [30]
    const float* kscale     = (const float*)d_in[6];  // [1]

    float* w_out     = (float*)d_out;                 // [4096,80]
    float* kappa_out = w_out + (size_t)B_SZ * CVOC;   // [4096,10]

    float* alpha_ws = (float*)d_ws;                   // [4096,10]
    float* beta_ws  = alpha_ws + (size_t)B_SZ * KMIX; // [4096,10]

    proj_act_kernel<<<B_SZ / 16, 64, 0, stream>>>(
        inputs, W, bias, prev_kappa, kscale, alpha_ws, beta_ws, kappa_out);

    phi_einsum_kernel<<<B_SZ, 256, 0, stream>>>(
        alpha_ws, beta_ws, kappa_out, onehot, seqlen, w_out);
}